// CoordinateDescentRouter_25391846654016
// MI455X (gfx1250) — compile-verified
//
#include <hip/hip_runtime.h>
#include <hip/hip_bf16.h>

// ---------------------------------------------------------------------------
// CoordinateDescentRouter for MI455X (gfx1250, wave32)
// Phase 1: s = x @ r  via V_WMMA_F32_16X16X4_F32, double-buffered LDS slices
//          streamed with GLOBAL_LOAD_ASYNC_TO_LDS_B128 (ASYNCcnt) when the
//          toolchain exposes the builtin (sync b128 staging as fallback).
// Phase 2: coor_descent (20 iters) + bitonic top-k, one workgroup per batch row
// Phase 3: routed-token gather (coalesced b128 copy)
// ---------------------------------------------------------------------------

typedef float v2f __attribute__((ext_vector_type(2)));
typedef float v8f __attribute__((ext_vector_type(8)));

#define B_SZ      4
#define N_TOK     8192
#define D_DIM     1024
#define K_MAX     1024
#define TILE_TOK  128
#define SLICE_D   32
#define LDS_STR   36            // padded row stride (floats): 16B aligned, bank-conflict-free
#define NSLICES   (D_DIM / SLICE_D)
#define LPT       4             // async b128 loads issued per thread per slice
#define MASK_VALUE (-3.402823466e38f)

#if __has_builtin(__builtin_amdgcn_global_load_async_to_lds_b128)
#define HAVE_ASYNC_LDS 1
// Exact parameter types per hipcc diagnostic: pointer to 16B int vector,
// global (AS1) source, LDS (AS3) destination.
typedef int b128_t __attribute__((vector_size(16)));
typedef __attribute__((address_space(1))) b128_t g_b128_t;
typedef __attribute__((address_space(3))) b128_t l_b128_t;
#if __has_builtin(__builtin_amdgcn_s_wait_asynccnt)
#define WAIT_ASYNC(n) __builtin_amdgcn_s_wait_asynccnt(n)
#else
#define WAIT_ASYNC(n) asm volatile("s_wait_asynccnt %0" ::"i"(n) : "memory")
#endif
#endif

// Stage one 128-token x 32-dim slice of x into an LDS buffer.
__device__ __forceinline__ void stage_slice(const float* __restrict__ xbase,
                                            int slice, float* dst, int tid) {
  const float* src = xbase + slice * SLICE_D;
#pragma unroll
  for (int i = tid; i < TILE_TOK * (SLICE_D / 4); i += 256) {   // 4 iters/thread
    const int tok = i >> 3;          // 8 float4 per token row
    const int c4  = (i & 7) * 4;
#ifdef HAVE_ASYNC_LDS
    __builtin_amdgcn_global_load_async_to_lds_b128(
        (g_b128_t*)(src + (size_t)tok * D_DIM + c4),
        (l_b128_t*)(dst + tok * LDS_STR + c4),
        /*offset=*/0, /*cpol=*/0);
#else
    const float4 v = *(const float4*)(src + (size_t)tok * D_DIM + c4);
    *(float4*)(dst + tok * LDS_STR + c4) = v;
#endif
  }
}

// ---------------------------------------------------------------------------
// Phase 1: scores via f32 WMMA.
// A-matrix (16x4): routing token replicated in every row (lanes 0-15: K0,K1;
// lanes 16-31: K2,K3). B-matrix (4x16): 16 token fragments (lane n<16 ->
// token n rows K0,K1; lane 16+n -> token n rows K2,K3). C row M=0 (VGPR0,
// lanes 0-15) yields the 16 dot products.
// ---------------------------------------------------------------------------
__global__ __launch_bounds__(256) void scores_wmma_kernel(
    const float* __restrict__ x, const float* __restrict__ rt,
    float* __restrict__ s_out) {
  __shared__ float rs[D_DIM];
  __shared__ float xs[2][TILE_TOK * LDS_STR];

  const int tid  = threadIdx.x;
  const int b    = blockIdx.y;
  const int tile = blockIdx.x * TILE_TOK;

  for (int i = tid; i < D_DIM; i += 256) rs[i] = rt[i];

  const int wave    = tid >> 5;
  const int lane    = tid & 31;
  const int lj      = lane & 15;
  const int half2   = (lane >> 4) * 2;   // 0: lanes 0-15 (K0,K1); 2: lanes 16-31 (K2,K3)
  const int tokBase = wave * 16;

  const float* xbase = x + ((size_t)b * N_TOK + tile) * D_DIM;

  v8f acc = {};
  stage_slice(xbase, 0, xs[0], tid);      // prologue: slice 0 in flight

  for (int cur = 0; cur < NSLICES; ++cur) {
    const int nxt = cur + 1;
    if (nxt < NSLICES) stage_slice(xbase, nxt, xs[nxt & 1], tid);
#ifdef HAVE_ASYNC_LDS
    // Async loads complete in order: <=LPT outstanding => slice `cur` landed.
    if (nxt < NSLICES) WAIT_ASYNC(LPT); else WAIT_ASYNC(0);
#endif
    __syncthreads();                      // slice `cur` visible to all waves

    const float* rsl  = rs + cur * SLICE_D;
    const float* xrow = &xs[cur & 1][(tokBase + lj) * LDS_STR];
#pragma unroll
    for (int k = 0; k < SLICE_D; k += 4) {
      v2f av = *(const v2f*)(rsl + k + half2);    // routing token fragment (A)
      v2f bv = *(const v2f*)(xrow + k + half2);   // token fragment (B)
      acc = __builtin_amdgcn_wmma_f32_16x16x4_f32(
          /*neg_a=*/false, av, /*neg_b=*/false, bv,
          /*c_mod=*/(short)0, acc, /*reuse_a=*/false, /*reuse_b=*/false);
    }
    __syncthreads();                      // all waves done before buffer reuse
  }

  // C layout: VGPR0 lanes 0-15 = row M=0, cols N=0..15.
  if (lane < 16)
    s_out[(size_t)b * N_TOK + tile + tokBase + lane] = acc[0];
}

// ---------------------------------------------------------------------------
// Reductions (wave32-correct) for the 1024-thread descent workgroup.
// ---------------------------------------------------------------------------
__device__ __forceinline__ float block_max_1024(float v, float* red) {
  const int lane = threadIdx.x & 31, w = threadIdx.x >> 5;
#pragma unroll
  for (int m = 16; m; m >>= 1) v = fmaxf(v, __shfl_xor(v, m, 32));
  __syncthreads();
  if (lane == 0) red[w] = v;
  __syncthreads();
  if (w == 0) {
    float t = red[lane];
#pragma unroll
    for (int m = 16; m; m >>= 1) t = fmaxf(t, __shfl_xor(t, m, 32));
    if (lane == 0) red[0] = t;
  }
  __syncthreads();
  return red[0];
}

__device__ __forceinline__ float block_sum_1024(float v, float* red) {
  const int lane = threadIdx.x & 31, w = threadIdx.x >> 5;
#pragma unroll
  for (int m = 16; m; m >>= 1) v += __shfl_xor(v, m, 32);
  __syncthreads();
  if (lane == 0) red[w] = v;
  __syncthreads();
  if (w == 0) {
    float t = red[lane];
#pragma unroll
    for (int m = 16; m; m >>= 1) t += __shfl_xor(t, m, 32);
    if (lane == 0) red[0] = t;
  }
  __syncthreads();
  return red[0];
}

// ---------------------------------------------------------------------------
// Phase 2: coordinate descent + bitonic top-k. One workgroup per batch row.
// s, b, mask live in registers (8 elements per thread, stride-1024 layout).
// Keys: (score_bits << 32) | (0xFFFFFFFF - index)  -> descending sort gives
// value-desc order with low-index tie-break, matching jax.lax.top_k.
// ---------------------------------------------------------------------------
__global__ __launch_bounds__(1024) void router_topk_kernel(
    const float* __restrict__ s_in, const unsigned char* __restrict__ mask,
    const int* __restrict__ num_tokens_p, float* __restrict__ out,
    int* __restrict__ idx_ws) {
  __shared__ unsigned long long keys[N_TOK];   // 64 KB
  float* red = (float*)keys;                   // reused as scratch pre-sort

  const int tid = threadIdx.x;
  const int b   = blockIdx.x;
  const int K   = *num_tokens_p;

  const float eff_k = fminf((float)K * (9.0f / 8.0f), (float)N_TOK);
  const float logk  = logf(fmaxf(eff_k, 1e-20f));

  float sv[8], bb[8];
  bool  mk[8];
#pragma unroll
  for (int e = 0; e < 8; ++e) {
    const int n = tid + e * 1024;
    mk[e] = mask[(size_t)b * N_TOK + n] != 0;
    const float sraw = s_in[(size_t)b * N_TOK + n];
    sv[e] = mk[e] ? sraw : MASK_VALUE;
    bb[e] = -sv[e];
  }

  float cur_eps = 4.0f, a = 0.0f;
  for (int it = 0; it < 20; ++it) {
    float sb[8];
    float lmax = MASK_VALUE;
#pragma unroll
    for (int e = 0; e < 8; ++e) {
      sb[e] = mk[e] ? (sv[e] + bb[e]) / cur_eps : MASK_VALUE;
      lmax  = fmaxf(lmax, sb[e]);
    }
    const float m = block_max_1024(lmax, red);
    float lsum = 0.0f;
#pragma unroll
    for (int e = 0; e < 8; ++e) lsum += expf(sb[e] - m);   // masked -> exp(-inf)=0
    const float ssum = block_sum_1024(lsum, red);
    const float lse  = m + logf(ssum);
    a = cur_eps * (logk - lse);
#pragma unroll
    for (int e = 0; e < 8; ++e) bb[e] = -fmaxf(sv[e] + a, 0.0f);
    cur_eps = fmaxf(cur_eps * 0.7f, 0.03f);
  }

  __syncthreads();   // red scratch done; keys can be overwritten
#pragma unroll
  for (int e = 0; e < 8; ++e) {
    const int n = tid + e * 1024;
    const float sc = mk[e] ? expf((sv[e] + a + bb[e]) / cur_eps) : 0.0f;
    keys[n] = ((unsigned long long)__float_as_uint(sc) << 32) |
              (unsigned long long)(0xFFFFFFFFu - (unsigned)n);
  }

  // Bitonic sort, descending.
  for (unsigned k = 2; k <= N_TOK; k <<= 1) {
    for (unsigned j = k >> 1; j > 0; j >>= 1) {
      __syncthreads();
      for (unsigned t = tid; t < N_TOK / 2; t += 1024) {
        const unsigned i = ((t & ~(j - 1)) << 1) | (t & (j - 1));
        const unsigned p = i | j;
        const bool larger_at_i = ((i & k) == 0);
        const unsigned long long ki = keys[i], kp = keys[p];
        if ((ki < kp) == larger_at_i) { keys[i] = kp; keys[p] = ki; }
      }
    }
  }
  __syncthreads();

  // Outputs: [indices | scores | tokens | mask], offsets computed with device K.
  const size_t BK = (size_t)B_SZ * (size_t)K;
  for (int i = tid; i < K; i += 1024) {
    const unsigned long long kk = keys[i];
    const unsigned idx = 0xFFFFFFFFu - (unsigned)(kk & 0xFFFFFFFFull);
    const bool md = mask[(size_t)b * N_TOK + idx] != 0;
    const float one = md ? 1.0f : 0.0f;   // straight-through: value is exactly 1
    out[(size_t)b * K + i]                               = (float)idx;  // indices
    out[BK + (size_t)b * K + i]                          = one;         // scores
    out[2 * BK + BK * (size_t)D_DIM + (size_t)b * K + i] = one;         // routed_mask
    idx_ws[b * K_MAX + i] = (int)idx;
  }
}

// ---------------------------------------------------------------------------
// Phase 3: gather routed tokens. One block per (token, batch); one float4/lane.
// ---------------------------------------------------------------------------
__global__ __launch_bounds__(256) void gather_kernel(
    const float* __restrict__ x, const int* __restrict__ idx_ws,
    const int* __restrict__ num_tokens_p, float* __restrict__ out) {
  const int K = *num_tokens_p;
  const int k = blockIdx.x;
  const int b = blockIdx.y;
  if (k >= K) return;
  const int idx = idx_ws[b * K_MAX + k];
  const float* src = x + ((size_t)b * N_TOK + idx) * D_DIM;
  const size_t BK = (size_t)B_SZ * (size_t)K;
  float* dst = out + 2 * BK + ((size_t)b * K + k) * (size_t)D_DIM;
  __builtin_prefetch(src + threadIdx.x * 4, 0, 1);     // global_prefetch_b8
  const float4 v = *(const float4*)(src + threadIdx.x * 4);
  *(float4*)(dst + threadIdx.x * 4) = v;
}

// ---------------------------------------------------------------------------
extern "C" void kernel_launch(void* const* d_in, const int* in_sizes, int n_in,
                              void* d_out, int out_size, void* d_ws, size_t ws_size,
                              hipStream_t stream) {
  const float*         x    = (const float*)d_in[0];          // (B, N, D) f32
  const float*         rt   = (const float*)d_in[1];          // (1, D)    f32
  const unsigned char* mask = (const unsigned char*)d_in[2];  // (B, N)    bool
  const int*           ntok = (const int*)d_in[3];            // scalar int
  float* out = (float*)d_out;

  float* s_ws   = (float*)d_ws;                                        // B*N f32
  int*   idx_ws = (int*)((char*)d_ws + (size_t)B_SZ * N_TOK * sizeof(float));

  scores_wmma_kernel<<<dim3(N_TOK / TILE_TOK, B_SZ), 256, 0, stream>>>(x, rt, s_ws);
  router_topk_kernel<<<dim3(B_SZ), 1024, 0, stream>>>(s_ws, mask, ntok, out, idx_ws);
  gather_kernel<<<dim3(K_MAX, B_SZ), 256, 0, stream>>>(x, idx_ws, ntok, out);
}